// relative_depth_crit_35579509080324
// MI455X (gfx1250) — compile-verified
//
#include <hip/hip_runtime.h>
#include <math.h>

// ---------------------------------------------------------------------------
// Relative-depth ordinal loss (gather + elementwise + mean) for MI455X/gfx1250
//
//   B=32 batches, HxW=1024x1024 fp32 depth, N=100000 point pairs per batch.
//   loss = mean over B*N of  mask*softplus(-gt*diff) + (1-mask)*diff^2
//
// Memory-bound: 64MB streamed indices (non-temporal) + 128MB depth working
// set that fits the 192MB L2. Reduction uses V_WMMA_F32_16X16X4_F32 with an
// all-ones B matrix as an exact 32-lane wave sum (layout-independent trick).
// ---------------------------------------------------------------------------

typedef __attribute__((ext_vector_type(2))) float v2f;
typedef __attribute__((ext_vector_type(8))) float v8f;
typedef __attribute__((ext_vector_type(4))) int   v4i;   // clang ext vector:
                                                         // valid for
                                                         // nontemporal builtin

#define BB 32
#define HH 1024
#define WW 1024
#define NN 100000
#define THREADS 256
#define BLOCKS_PER_BATCH 64
#define NPARTIAL (BB * BLOCKS_PER_BATCH)   /* 2048 block partial sums */

// Exact sum of (x + y) over all 32 lanes of the wave, returned in every lane.
// A-matrix (16x4 f32, 2 VGPRs/lane): lane L<16 holds A[L][0],A[L][1];
// lane L>=16 holds A[L-16][2],A[L-16][3].  B = all ones (layout independent).
// => D[m][n] = (x+y)_m + (x+y)_{m+16} for every n, so sum of the 8 D VGPRs
// plus one shfl_xor(16) yields the full wave sum.  Requires EXEC == all 1s.
__device__ __forceinline__ float wave_sum_wmma(float x, float y) {
  v2f a;    a.x = x;    a.y = y;
  v2f ones; ones.x = 1.0f; ones.y = 1.0f;
  v8f c = {};
  c = __builtin_amdgcn_wmma_f32_16x16x4_f32(false, a, false, ones,
                                            (short)0, c, false, false);
  float s = ((c[0] + c[1]) + (c[2] + c[3])) + ((c[4] + c[5]) + (c[6] + c[7]));
  s += __shfl_xor(s, 16, 32);
  return s;
}

__device__ __forceinline__ float point_loss(const float* __restrict__ db,
                                            int xa, int ya, int xb, int yb,
                                            int o) {
  float zA = db[(size_t)ya * WW + xa];
  float zB = db[(size_t)yb * WW + xb];
  float diff = zA - zB;
  float gt   = (float)(o - 1);           // -1, 0, +1
  float t    = -gt * diff;
  // numerically stable softplus(t) = max(t,0) + log1p(exp(-|t|))
  float sp   = fmaxf(t, 0.0f) + log1pf(expf(-fabsf(t)));
  return (o == 1) ? diff * diff : sp;    // mask = |gt|
}

__global__ __launch_bounds__(THREADS)
void rdepth_partial_kernel(const float* __restrict__ depth,
                           const int* __restrict__ xA,
                           const int* __restrict__ yA,
                           const int* __restrict__ xB,
                           const int* __restrict__ yB,
                           const int* __restrict__ ord,
                           float* __restrict__ partial) {
  const int b = blockIdx.y;
  const float* __restrict__ db = depth + (size_t)b * (HH * WW);
  const int base = b * NN;               // N % 4 == 0 -> 16B aligned chunks

  const v4i* __restrict__ vxA = reinterpret_cast<const v4i*>(xA + base);
  const v4i* __restrict__ vyA = reinterpret_cast<const v4i*>(yA + base);
  const v4i* __restrict__ vxB = reinterpret_cast<const v4i*>(xB + base);
  const v4i* __restrict__ vyB = reinterpret_cast<const v4i*>(yB + base);
  const v4i* __restrict__ vod = reinterpret_cast<const v4i*>(ord + base);

  const int tid    = blockIdx.x * THREADS + threadIdx.x;
  const int stride = BLOCKS_PER_BATCH * THREADS;
  const int nchunk = NN / 4;

  float a0 = 0.0f, a1 = 0.0f;            // two accumulators for ILP
  for (int c = tid; c < nchunk; c += stride) {
    // index streams have zero reuse: non-temporal so 192MB L2 keeps depth
    v4i ax = __builtin_nontemporal_load(&vxA[c]);
    v4i ay = __builtin_nontemporal_load(&vyA[c]);
    v4i bx = __builtin_nontemporal_load(&vxB[c]);
    v4i by = __builtin_nontemporal_load(&vyB[c]);
    v4i od = __builtin_nontemporal_load(&vod[c]);
    a0 += point_loss(db, ax.x, ay.x, bx.x, by.x, od.x);
    a1 += point_loss(db, ax.y, ay.y, bx.y, by.y, od.y);
    a0 += point_loss(db, ax.z, ay.z, bx.z, by.z, od.z);
    a1 += point_loss(db, ax.w, ay.w, bx.w, by.w, od.w);
  }

  // wave32 reduction via WMMA (all lanes active, full 256-thread blocks)
  float wsum = wave_sum_wmma(a0, a1);

  __shared__ float lds[THREADS / 32];
  const int lane = threadIdx.x & 31;
  const int wv   = threadIdx.x >> 5;
  if (lane == 0) lds[wv] = wsum;
  __syncthreads();
  if (threadIdx.x == 0) {
    float s = 0.0f;
#pragma unroll
    for (int w = 0; w < THREADS / 32; ++w) s += lds[w];
    partial[blockIdx.y * gridDim.x + blockIdx.x] = s;  // every slot written
  }
}

__global__ __launch_bounds__(THREADS)
void rdepth_final_kernel(const float* __restrict__ partial,
                         float* __restrict__ out) {
  float a0 = 0.0f, a1 = 0.0f;
  // NPARTIAL = 2048 = 4 * (2*THREADS): fixed-order deterministic accumulation
  for (int i = threadIdx.x; i < NPARTIAL; i += 2 * THREADS) {
    a0 += partial[i];
    a1 += partial[i + THREADS];
  }
  float wsum = wave_sum_wmma(a0, a1);

  __shared__ float lds[THREADS / 32];
  const int lane = threadIdx.x & 31;
  const int wv   = threadIdx.x >> 5;
  if (lane == 0) lds[wv] = wsum;
  __syncthreads();
  if (threadIdx.x == 0) {
    float s = 0.0f;
#pragma unroll
    for (int w = 0; w < THREADS / 32; ++w) s += lds[w];
    out[0] = s * (1.0f / (float)(BB * NN));  // mean over 3.2M points
  }
}

extern "C" void kernel_launch(void* const* d_in, const int* in_sizes, int n_in,
                              void* d_out, int out_size, void* d_ws,
                              size_t ws_size, hipStream_t stream) {
  (void)in_sizes; (void)n_in; (void)out_size; (void)ws_size;
  const float* depth = (const float*)d_in[0];
  const int*   xA    = (const int*)d_in[1];
  const int*   yA    = (const int*)d_in[2];
  const int*   xB    = (const int*)d_in[3];
  const int*   yB    = (const int*)d_in[4];
  const int*   ord   = (const int*)d_in[5];
  float* partial = (float*)d_ws;         // 2048 floats = 8KB scratch
  float* out     = (float*)d_out;

  dim3 grid(BLOCKS_PER_BATCH, BB);
  rdepth_partial_kernel<<<grid, THREADS, 0, stream>>>(depth, xA, yA, xB, yB,
                                                      ord, partial);
  rdepth_final_kernel<<<1, THREADS, 0, stream>>>(partial, out);
}